// AdaptiveFlowRouter_53369263620435
// MI455X (gfx1250) — compile-verified
//
#include <hip/hip_runtime.h>
#include <hip/hip_bf16.h>

// ---------------------------------------------------------------------------
// AdaptiveFlowRouter for gfx1250 (MI455X)
//   kernel 1: per-token softmax(x@Wp), sigmoid(x@Wi); stage w*intensity (A
//             matrix for WMMA), raw w, intensity, per-token entropy in ws.
//   kernel 2: deterministic reduction -> entropy mean / intensity mean /
//             diversity (std ddof=1 of pattern means) into d_out tail.
//   kernel 3: per 16-token group: flow tiles via V_WMMA_F32_16X16X4_F32,
//             exact 31-bit radix select (4 x 256-bin LDS histogram passes,
//             flow recomputed each pass from L2-hot patterns), then a write
//             pass with non-temporal stores (268MB stream, keep L2 for
//             patterns).
// ---------------------------------------------------------------------------

typedef float v2f_t __attribute__((ext_vector_type(2)));
typedef float v8f_t __attribute__((ext_vector_type(8)));

#define IN_DIM     256
#define OUT_DIM    256
#define NPAT       8
#define NTOK       1024            // B*S
#define ELEMS      65536           // OUT_DIM*IN_DIM per token
#define TILES      4096            // ELEMS / 16
#define K_KEEP     6553u           // int(ELEMS * 0.1)
#define NWAVES     32              // 1024-thread block, wave32

// ws layout (floats)
#define WS_A       0               // [1024][8]  w * intensity  (WMMA A)
#define WS_W       8192            // [1024][8]  raw softmax weights
#define WS_I       16384           // [1024]     intensity
#define WS_E       17408           // [1024]     per-token entropy

// ---------------------------------------------------------------------------
// Kernel 1: weights / intensity / per-token entropy. 1 token per wave32.
// ---------------------------------------------------------------------------
__global__ void __launch_bounds__(256)
k_weights(const float* __restrict__ x,  const float* __restrict__ Wp,
          const float* __restrict__ bp, const float* __restrict__ Wi,
          const float* __restrict__ bi, float* __restrict__ ws) {
  const int tok  = blockIdx.x * 8 + (threadIdx.x >> 5);
  const int lane = threadIdx.x & 31;
  const float* xt = x + (size_t)tok * IN_DIM;

  float acc[9];
#pragma unroll
  for (int c = 0; c < 9; ++c) acc[c] = 0.0f;

#pragma unroll
  for (int j = 0; j < 8; ++j) {
    const int i  = lane + j * 32;
    const float xv = xt[i];
#pragma unroll
    for (int c = 0; c < 8; ++c) acc[c] += xv * Wp[i * NPAT + c];   // Wp: [IN, P]
    acc[8] += xv * Wi[i];                                          // Wi: [IN, 1]
  }
  // wave32 xor-reduction (all lanes end with the sum)
#pragma unroll
  for (int c = 0; c < 9; ++c) {
#pragma unroll
    for (int off = 16; off > 0; off >>= 1)
      acc[c] += __shfl_xor(acc[c], off, 32);
  }

  if (lane == 0) {
    float lg[8], mx = -3.4e38f;
#pragma unroll
    for (int c = 0; c < 8; ++c) { lg[c] = acc[c] + bp[c]; mx = fmaxf(mx, lg[c]); }
    float e[8], s = 0.0f;
#pragma unroll
    for (int c = 0; c < 8; ++c) { e[c] = __expf(lg[c] - mx); s += e[c]; }
    const float inv   = 1.0f / s;
    const float inten = 1.0f / (1.0f + __expf(-(acc[8] + bi[0])));
    float ent = 0.0f;
#pragma unroll
    for (int c = 0; c < 8; ++c) {
      const float w = e[c] * inv;
      ws[WS_W + tok * NPAT + c] = w;
      ws[WS_A + tok * NPAT + c] = w * inten;   // fold intensity into WMMA A
      ent -= w * __logf(w + 1e-8f);
    }
    ws[WS_I + tok] = inten;
    ws[WS_E + tok] = ent;
  }
}

// ---------------------------------------------------------------------------
// Kernel 2: metric reductions (deterministic tree reduce, single block).
// ---------------------------------------------------------------------------
__global__ void __launch_bounds__(256)
k_metrics(const float* __restrict__ ws, float* __restrict__ out3) {
  __shared__ float red[256];
  __shared__ float mp[8];
  const int tid = threadIdx.x;

  // entropy mean
  float s = 0.0f;
  for (int t = tid; t < NTOK; t += 256) s += ws[WS_E + t];
  red[tid] = s; __syncthreads();
  for (int off = 128; off > 0; off >>= 1) {
    if (tid < off) red[tid] += red[tid + off];
    __syncthreads();
  }
  if (tid == 0) out3[0] = red[0] * (1.0f / NTOK);
  __syncthreads();

  // intensity mean
  s = 0.0f;
  for (int t = tid; t < NTOK; t += 256) s += ws[WS_I + t];
  red[tid] = s; __syncthreads();
  for (int off = 128; off > 0; off >>= 1) {
    if (tid < off) red[tid] += red[tid + off];
    __syncthreads();
  }
  if (tid == 0) out3[1] = red[0] * (1.0f / NTOK);
  __syncthreads();

  // per-pattern means -> std (ddof=1)
  for (int p = 0; p < NPAT; ++p) {
    s = 0.0f;
    for (int t = tid; t < NTOK; t += 256) s += ws[WS_W + t * NPAT + p];
    red[tid] = s; __syncthreads();
    for (int off = 128; off > 0; off >>= 1) {
      if (tid < off) red[tid] += red[tid + off];
      __syncthreads();
    }
    if (tid == 0) mp[p] = red[0] * (1.0f / NTOK);
    __syncthreads();
  }
  if (tid == 0) {
    float mean = 0.0f;
#pragma unroll
    for (int p = 0; p < 8; ++p) mean += mp[p];
    mean *= 0.125f;
    float var = 0.0f;
#pragma unroll
    for (int p = 0; p < 8; ++p) { const float d = mp[p] - mean; var += d * d; }
    out3[2] = sqrtf(var / 7.0f);
  }
}

// ---------------------------------------------------------------------------
// Flow tile: D[16 tok,16 elem] = A[16,8] x B[8,16] via two f32 K=4 WMMAs.
// Identical instruction sequence every pass -> bitwise-identical flow values.
// ---------------------------------------------------------------------------
__device__ __forceinline__ v8f_t
flow_tile(const float* __restrict__ P, int ebase, int n, int kg,
          v2f_t a_lo, v2f_t a_hi) {
  const float* pe = P + ebase + n;
  v2f_t b_lo, b_hi;                       // B[K][N]: K = kg + vgpr, N = lane&15
  b_lo.x = pe[(kg + 0) * ELEMS];
  b_lo.y = pe[(kg + 1) * ELEMS];
  b_hi.x = pe[(kg + 4) * ELEMS];
  b_hi.y = pe[(kg + 5) * ELEMS];
  v8f_t c = {};
  c = __builtin_amdgcn_wmma_f32_16x16x4_f32(false, a_lo, false, b_lo,
                                            (short)0, c, false, false);
  c = __builtin_amdgcn_wmma_f32_16x16x4_f32(false, a_hi, false, b_hi,
                                            (short)0, c, false, false);
  return c;
}

// ---------------------------------------------------------------------------
// Kernel 3: flow + exact top-k threshold + sparse write. 16 tokens / block.
// ---------------------------------------------------------------------------
__global__ void __launch_bounds__(1024)
k_flow(const float* __restrict__ P, const float* __restrict__ ws,
       float* __restrict__ out) {
  __shared__ unsigned hist[16][256];
  __shared__ unsigned prefix[16];
  __shared__ unsigned kneed[16];

  const int t0   = blockIdx.x * 16;
  const int tid  = threadIdx.x;
  const int lane = tid & 31;
  // wave id is wave-uniform: force scalar so tile loops get s_cmp/s_cbranch
  // control flow (EXEC provably all-1s at every WMMA).
  const int wave = __builtin_amdgcn_readfirstlane(tid >> 5);
  const int n    = lane & 15;        // N (elem within tile) / A row M
  const int hi   = lane >> 4;        // half-wave
  const int kg   = hi << 1;          // K pair base (0 or 2)
  const int hi8  = hi << 3;          // D row offset (0 or 8)

  // A[16,8]: M = lane&15 (token), a.x->K=kg, a.y->K=kg+1 (+4 for 2nd WMMA)
  const float* Ar = ws + WS_A + (size_t)(t0 + n) * NPAT;
  v2f_t a_lo, a_hi;
  a_lo.x = Ar[kg];     a_lo.y = Ar[kg + 1];
  a_hi.x = Ar[kg + 4]; a_hi.y = Ar[kg + 5];

  if (tid < 16) kneed[tid] = K_KEEP;
  __syncthreads();

  const int shifts[4] = {23, 15, 7, 0};   // 8+8+8+7 bits of |v| pattern
  unsigned pfx[8];

  for (int level = 0; level < 4; ++level) {
    {  // clear histogram: 4 consecutive dwords/thread -> ds_store_b128
      unsigned* h = &hist[0][0] + tid * 4;
#pragma unroll
      for (int j = 0; j < 4; ++j) h[j] = 0u;
    }
    __syncthreads();
    if (level > 0) {
#pragma unroll
      for (int r = 0; r < 8; ++r) pfx[r] = prefix[r + hi8];
    }
    const int      sh   = shifts[level];
    const int      psh  = level ? shifts[level - 1] : 31;
    const unsigned mask = (level == 3) ? 0x7Fu : 0xFFu;

#pragma unroll 2
    for (int tile = wave; tile < TILES; tile += NWAVES) {
      const v8f_t c = flow_tile(P, tile << 4, n, kg, a_lo, a_hi);
#pragma unroll
      for (int r = 0; r < 8; ++r) {
        const unsigned u = __float_as_uint(c[r]) & 0x7fffffffu;
        if (level == 0 || (u >> psh) == pfx[r])
          atomicAdd(&hist[r + hi8][(u >> sh) & mask], 1u);
      }
    }
    __syncthreads();

    if (tid < 16) {                       // per-token boundary-bin scan
      const unsigned need = kneed[tid];
      unsigned cum = 0; int b = 0;
      for (int i = 255; i >= 0; --i) {
        cum += hist[tid][i];
        if (cum >= need) { b = i; break; }
      }
      const unsigned above = cum - hist[tid][b];
      kneed[tid]  = need - above;
      prefix[tid] = (level == 0)
                      ? (unsigned)b
                      : ((prefix[tid] << (unsigned)(psh - sh)) | (unsigned)b);
    }
    __syncthreads();
  }

  // write pass: keep |v| >= threshold (prefix == exact 31-bit k-th value)
  unsigned th[8];
#pragma unroll
  for (int r = 0; r < 8; ++r) th[r] = prefix[r + hi8];

#pragma unroll 2
  for (int tile = wave; tile < TILES; tile += NWAVES) {
    const int ebase = tile << 4;
    const v8f_t c = flow_tile(P, ebase, n, kg, a_lo, a_hi);
#pragma unroll
    for (int r = 0; r < 8; ++r) {
      const unsigned u = __float_as_uint(c[r]) & 0x7fffffffu;
      const float v = (u >= th[r]) ? c[r] : 0.0f;
      __builtin_nontemporal_store(
          v, out + (size_t)(t0 + r + hi8) * ELEMS + (ebase + n));
    }
  }
}

// ---------------------------------------------------------------------------
extern "C" void kernel_launch(void* const* d_in, const int* in_sizes, int n_in,
                              void* d_out, int out_size, void* d_ws,
                              size_t ws_size, hipStream_t stream) {
  const float* x        = (const float*)d_in[0];
  const float* patterns = (const float*)d_in[1];
  const float* Wp       = (const float*)d_in[2];
  const float* bp       = (const float*)d_in[3];
  const float* Wi       = (const float*)d_in[4];
  const float* bi       = (const float*)d_in[5];
  float* out = (float*)d_out;
  float* ws  = (float*)d_ws;

  k_weights<<<NTOK / 8, 256, 0, stream>>>(x, Wp, bp, Wi, bi, ws);
  k_metrics<<<1, 256, 0, stream>>>(ws, out + (size_t)NTOK * ELEMS);
  k_flow<<<NTOK / 16, 1024, 0, stream>>>(patterns, ws, out);
}